// KANGroup1D_4037269258307
// MI455X (gfx1250) — compile-verified
//
#include <hip/hip_runtime.h>
#include <cstdint>

// Problem geometry (fixed by the reference)
#define C_CH    192
#define K_BINS  32
#define HW      16384          // H*W = 128*128
#define THREADS 256
#define VEC     4              // float4 per thread
#define BLOCKS_PER_PLANE (HW / (THREADS * VEC))   // 16

__global__ __launch_bounds__(THREADS)
void kan_group1d_kernel(const float* __restrict__ x,
                        const float* __restrict__ alpha,
                        const float* __restrict__ a,
                        const float* __restrict__ b,
                        const float* __restrict__ id_gain,
                        const float* __restrict__ bias,
                        const int*   __restrict__ group_idx,
                        float* __restrict__ out)
{
    __shared__ float s_alpha[K_BINS];

    const int plane = blockIdx.x / BLOCKS_PER_PLANE;   // (b*C + c)
    const int chunk = blockIdx.x % BLOCKS_PER_PLANE;
    const int c     = plane % C_CH;

    // Block-uniform per-channel parameters -> scalar registers (s_load)
    const float A  = a[c];
    const float Bv = b[c];
    const float Gn = id_gain[c];
    const float Z  = bias[c];
    const int   g  = group_idx[c];

    // --- CDNA5 async path: DMA the 32-entry alpha row HBM -> LDS (no VGPR round trip).
    // Wave 0 (lanes 0..31, wave32) issues one async-to-LDS b32 per lane = 128 bytes.
    if (threadIdx.x < K_BINS) {
        unsigned lds_off = (unsigned)(uintptr_t)(&s_alpha[threadIdx.x]); // low 32b of flat LDS addr = LDS offset
        unsigned voff    = threadIdx.x * 4u;
        uint64_t srow    = (uint64_t)(uintptr_t)(alpha + (size_t)g * K_BINS);
        asm volatile(
            "global_load_async_to_lds_b32 %0, %1, %2\n\t"
            "s_wait_asynccnt 0"
            :: "v"(lds_off), "v"(voff), "s"(srow)
            : "memory");
    }
    __syncthreads();

    // lane l holds alpha[g][l]; gathers use the wave crossbar (ds_bpermute_b32, K==wave32)
    const int lane  = threadIdx.x & 31;
    const int avali = __float_as_int(s_alpha[lane]);

    const size_t base = (size_t)plane * HW
                      + (size_t)chunk * (THREADS * VEC)
                      + (size_t)threadIdx.x * VEC;

    const float4 xv = *(const float4*)(x + base);
    float xs[VEC] = {xv.x, xv.y, xv.z, xv.w};
    float r[VEC];

#pragma unroll
    for (int e = 0; e < VEC; ++e) {
        const float xe = xs[e];
        // affine + clamp to [-1.5, 1.5]
        float xa = fmaf(xe, A, Bv);
        xa = fminf(fmaxf(xa, -1.5f), 1.5f);
        // bin coordinate u in [-7.75, 38.75]
        const float u  = (xa + 1.0f) * 15.5f;
        const float fi = floorf(u);
        const float t  = u - fi;               // in [0,1)
        const int   i4 = ((int)fi) << 2;       // byte-scaled bin index
        // clamped byte offsets into the 32-entry row (v_med3_i32), feed bpermute directly
        const int j0 = min(max(i4 - 4, 0), (K_BINS - 1) * 4);
        const int j1 = min(max(i4,     0), (K_BINS - 1) * 4);
        const int j2 = min(max(i4 + 4, 0), (K_BINS - 1) * 4);
        const int j3 = min(max(i4 + 8, 0), (K_BINS - 1) * 4);
        const float a0 = __int_as_float(__builtin_amdgcn_ds_bpermute(j0, avali));
        const float a1 = __int_as_float(__builtin_amdgcn_ds_bpermute(j1, avali));
        const float a2 = __int_as_float(__builtin_amdgcn_ds_bpermute(j2, avali));
        const float a3 = __int_as_float(__builtin_amdgcn_ds_bpermute(j3, avali));
        // cubic basis, scaled by 6: w0=(1-t)^3, w1=4-6t^2+3t^3, w3=t^3, w2=6-w0-w1-w3
        const float t2 = t * t;
        const float t3 = t2 * t;
        const float s1 = 1.0f - t;
        const float w0 = s1 * s1 * s1;
        const float w1 = fmaf(3.0f, t3, fmaf(-6.0f, t2, 4.0f));
        const float w3 = t3;
        const float w2 = 6.0f - w0 - w1 - w3;
        float sp = a0 * w0;
        sp = fmaf(a1, w1, sp);
        sp = fmaf(a2, w2, sp);
        sp = fmaf(a3, w3, sp);
        // out = id_gain*x + spline/6 + bias
        r[e] = fmaf(Gn, xe, fmaf(sp, 0.16666667f, Z));
    }

    *(float4*)(out + base) = make_float4(r[0], r[1], r[2], r[3]);
}

extern "C" void kernel_launch(void* const* d_in, const int* in_sizes, int n_in,
                              void* d_out, int out_size, void* d_ws, size_t ws_size,
                              hipStream_t stream)
{
    const float* x      = (const float*)d_in[0];
    const float* alpha  = (const float*)d_in[1];
    const float* a      = (const float*)d_in[2];
    const float* b      = (const float*)d_in[3];
    const float* idg    = (const float*)d_in[4];
    const float* bias   = (const float*)d_in[5];
    const int*   gidx   = (const int*)d_in[6];
    float*       out    = (float*)d_out;

    const int n      = in_sizes[0];                 // B*C*H*W = 50331648
    const int planes = n / HW;                      // 3072
    const int grid   = planes * BLOCKS_PER_PLANE;   // 49152

    kan_group1d_kernel<<<grid, THREADS, 0, stream>>>(x, alpha, a, b, idg, bias, gidx, out);
}